// VariationalGCNEncoder_6030134083624
// MI455X (gfx1250) — compile-verified
//
#include <hip/hip_runtime.h>
#include <hip/hip_bf16.h>

typedef __attribute__((ext_vector_type(2))) float v2f;
typedef __attribute__((ext_vector_type(8))) float v8f;

#define BN_EPS 1e-5f

// ---- monotone float<->ordered-uint mapping for atomic segment-max ----
__device__ __forceinline__ unsigned f2o(float f) {
  unsigned u = __float_as_uint(f);
  return (u & 0x80000000u) ? ~u : (u | 0x80000000u);
}
__device__ __forceinline__ float o2f(unsigned u) {
  return (u & 0x80000000u) ? __uint_as_float(u & 0x7fffffffu) : __uint_as_float(~u);
}

__device__ __forceinline__ float lrelu02(float v) { return v > 0.f ? v : 0.2f * v; }
__device__ __forceinline__ float lrelu001(float v) { return v > 0.f ? v : 0.01f * v; }

// ---------------- zero init (contiguous scratch region) ----------------
__global__ void k_zero(float* p, size_t n) {
  size_t i = (size_t)blockIdx.x * blockDim.x + threadIdx.x;
  size_t stride = (size_t)gridDim.x * blockDim.x;
  for (; i < n; i += stride) p[i] = 0.f;
}

// ---------------- GAT linear: h[N,48] = x[N,32] @ W_gat[32,48] (WMMA f32) ----------------
__global__ void k_gat_gemm(const float* __restrict__ x, const float* __restrict__ Wg,
                           float* __restrict__ h, int N) {
  int lane = threadIdx.x & 31, wave = threadIdx.x >> 5;
  int hf = lane >> 4, l16 = lane & 15;
  int rowBase = (blockIdx.x * 8 + wave) * 16;
  v8f c0 = {}, c1 = {}, c2 = {};
#pragma unroll
  for (int k0 = 0; k0 < 32; k0 += 4) {
    int ka = k0 + 2 * hf;                       // A layout: lanes>=16 hold K=g+2
    v2f a = *(const v2f*)(x + (size_t)(rowBase + l16) * 32 + ka);
    const float* w0 = Wg + ka * 48;             // B layout: lane holds (K=ka,N=l16),(K=ka+1,N=l16)
    v2f bb0, bb1, bb2;
    bb0[0] = w0[l16];      bb0[1] = w0[48 + l16];
    bb1[0] = w0[16 + l16]; bb1[1] = w0[64 + l16];
    bb2[0] = w0[32 + l16]; bb2[1] = w0[80 + l16];
    c0 = __builtin_amdgcn_wmma_f32_16x16x4_f32(false, a, false, bb0, (short)0, c0, false, false);
    c1 = __builtin_amdgcn_wmma_f32_16x16x4_f32(false, a, false, bb1, (short)0, c1, false, false);
    c2 = __builtin_amdgcn_wmma_f32_16x16x4_f32(false, a, false, bb2, (short)0, c2, false, false);
  }
#pragma unroll
  for (int g = 0; g < 8; ++g) {                 // D layout: row = g + 8*half, col = l16
    size_t r = (size_t)(rowBase + g + 8 * hf);
    h[r * 48 + l16]      = c0[g];
    h[r * 48 + 16 + l16] = c1[g];
    h[r * 48 + 32 + l16] = c2[g];
  }
}

// ---------------- attention coefficients a_src/a_dst [N,3] ----------------
__global__ void k_att(const float* __restrict__ h, const float* __restrict__ attS,
                      const float* __restrict__ attD, float* __restrict__ asrc,
                      float* __restrict__ adst, int N) {
  int n = blockIdx.x * blockDim.x + threadIdx.x;
  if (n >= N) return;
  const float* hn = h + (size_t)n * 48;
#pragma unroll
  for (int t = 0; t < 3; ++t) {
    float as = 0.f, ad = 0.f;
#pragma unroll
    for (int d = 0; d < 16; ++d) {
      float hv = hn[t * 16 + d];
      as += hv * attS[t * 16 + d];
      ad += hv * attD[t * 16 + d];
    }
    asrc[(size_t)n * 3 + t] = as;
    adst[(size_t)n * 3 + t] = ad;
  }
}

// ---------------- edge pass 1: segment max (ordered-int atomicMax) ----------------
__global__ void k_edge_max(const int* __restrict__ ei, const float* __restrict__ asrc,
                           const float* __restrict__ adst, unsigned* __restrict__ mord,
                           long long E, long long Etot) {
  long long e = (long long)blockIdx.x * blockDim.x + threadIdx.x;
  if (e >= Etot) return;
  int s, d;
  if (e < E) { s = ei[e]; d = ei[E + e]; } else { s = d = (int)(e - E); }
#pragma unroll
  for (int t = 0; t < 3; ++t) {
    float v = lrelu02(asrc[(size_t)s * 3 + t] + adst[(size_t)d * 3 + t]);
    atomicMax(&mord[(size_t)d * 3 + t], f2o(v));
  }
}

// ---------------- edge pass 2: softmax denominator + degree ----------------
__global__ void k_edge_z(const int* __restrict__ ei, const float* __restrict__ asrc,
                         const float* __restrict__ adst, const unsigned* __restrict__ mord,
                         float* __restrict__ z, float* __restrict__ deg,
                         long long E, long long Etot) {
  long long e = (long long)blockIdx.x * blockDim.x + threadIdx.x;
  if (e >= Etot) return;
  int s, d;
  if (e < E) { s = ei[e]; d = ei[E + e]; } else { s = d = (int)(e - E); }
#pragma unroll
  for (int t = 0; t < 3; ++t) {
    float v = lrelu02(asrc[(size_t)s * 3 + t] + adst[(size_t)d * 3 + t]);
    float m = o2f(mord[(size_t)d * 3 + t]);
    atomicAdd(&z[(size_t)d * 3 + t], __expf(v - m));
  }
  atomicAdd(&deg[d], 1.0f);
}

// ---------------- edge pass 3: weighted message aggregation ----------------
__global__ void k_edge_msg(const int* __restrict__ ei, const float* __restrict__ asrc,
                           const float* __restrict__ adst, const unsigned* __restrict__ mord,
                           const float* __restrict__ z, const float* __restrict__ h,
                           float* __restrict__ out1, long long E, long long Etot) {
  long long e = (long long)blockIdx.x * blockDim.x + threadIdx.x;
  if (e >= Etot) return;
  int s, d;
  if (e < E) { s = ei[e]; d = ei[E + e]; } else { s = d = (int)(e - E); }
#pragma unroll
  for (int t = 0; t < 3; ++t) {
    float v = lrelu02(asrc[(size_t)s * 3 + t] + adst[(size_t)d * 3 + t]);
    float m = o2f(mord[(size_t)d * 3 + t]);
    float alpha = __expf(v - m) / z[(size_t)d * 3 + t];
    const float* hs = h + (size_t)s * 48 + t * 16;
    float* od = out1 + (size_t)d * 48 + t * 16;
#pragma unroll
    for (int dd = 0; dd < 16; ++dd) atomicAdd(&od[dd], hs[dd] * alpha);
  }
}

// -------- head transform: ht[N,16] = out1[N,48] @ W_ht[48,16] (WMMA) + BN1 stats --------
__global__ void k_ht_gemm(const float* __restrict__ out1, const float* __restrict__ Wht,
                          float* __restrict__ ht, float* __restrict__ stat1, int N) {
  int lane = threadIdx.x & 31, wave = threadIdx.x >> 5;
  int hf = lane >> 4, l16 = lane & 15;
  int rowBase = (blockIdx.x * 8 + wave) * 16;
  v8f c = {};
#pragma unroll
  for (int k0 = 0; k0 < 48; k0 += 4) {
    int ka = k0 + 2 * hf;
    v2f a = *(const v2f*)(out1 + (size_t)(rowBase + l16) * 48 + ka);
    v2f b;
    b[0] = Wht[ka * 16 + l16];
    b[1] = Wht[(ka + 1) * 16 + l16];
    c = __builtin_amdgcn_wmma_f32_16x16x4_f32(false, a, false, b, (short)0, c, false, false);
  }
  float s = 0.f, q = 0.f;
#pragma unroll
  for (int g = 0; g < 8; ++g) {
    size_t r = (size_t)(rowBase + g + 8 * hf);
    ht[r * 16 + l16] = c[g];
    s += c[g];
    q += c[g] * c[g];
  }
  // column-wise partials: lanes l and l+16 share a column, different rows
  s += __shfl_xor(s, 16, 32);
  q += __shfl_xor(q, 16, 32);
  if (hf == 0) {
    atomicAdd(&stat1[l16], s);
    atomicAdd(&stat1[16 + l16], q);
  }
}

// ---- fold BN1 into GCN weight: Wg2[k][j]=s1[k]*Wgcn[k][j], c0[j]=sum t1[k]*Wgcn[k][j] ----
__global__ void k_fold1(const float* __restrict__ stat1, const float* __restrict__ g1,
                        const float* __restrict__ be1, const float* __restrict__ Wgcn,
                        float* __restrict__ fold1, int N) {
  __shared__ float s1[16], t1[16];
  int t = threadIdx.x;
  if (t < 16) {
    float mean = stat1[t] / (float)N;
    float var = stat1[16 + t] / (float)N - mean * mean;
    float s = g1[t] * rsqrtf(var + BN_EPS);
    s1[t] = s;
    t1[t] = be1[t] - mean * s;
  }
  __syncthreads();
  if (t < 128) { int k = t >> 3, j = t & 7; fold1[t] = s1[k] * Wgcn[k * 8 + j]; }
  if (t < 8) {
    float c = 0.f;
    for (int k = 0; k < 16; ++k) c += t1[k] * Wgcn[k * 8 + t];
    fold1[128 + t] = c;
  }
}

// ---------------- xw[N,8] = bn1(ht) @ W_gcn ; dinv ; zero agg ----------------
__global__ void k_xw(const float* __restrict__ ht, const float* __restrict__ fold1,
                     const float* __restrict__ deg, float* __restrict__ xw,
                     float* __restrict__ dinv, float* __restrict__ agg, int N) {
  __shared__ float W[128], C0[8];
  int tid = threadIdx.x;
  if (tid < 128) W[tid] = fold1[tid];
  if (tid < 8) C0[tid] = fold1[128 + tid];
  __syncthreads();
  int n = blockIdx.x * blockDim.x + tid;
  if (n >= N) return;
  float acc[8];
#pragma unroll
  for (int j = 0; j < 8; ++j) acc[j] = C0[j];
#pragma unroll
  for (int k = 0; k < 16; ++k) {
    float v = ht[(size_t)n * 16 + k];
#pragma unroll
    for (int j = 0; j < 8; ++j) acc[j] += v * W[k * 8 + j];
  }
#pragma unroll
  for (int j = 0; j < 8; ++j) {
    xw[(size_t)n * 8 + j] = acc[j];
    agg[(size_t)n * 8 + j] = 0.f;   // re-zero aliased accumulator for GCN pass
  }
  dinv[n] = rsqrtf(deg[n]);
}

// ---------------- GCN edge pass: agg[dst] += xw[src] * dinv[src]*dinv[dst] ----------------
__global__ void k_gcn_edge(const int* __restrict__ ei, const float* __restrict__ dinv,
                           const float* __restrict__ xw, float* __restrict__ agg,
                           long long E, long long Etot) {
  long long e = (long long)blockIdx.x * blockDim.x + threadIdx.x;
  if (e >= Etot) return;
  int s, d;
  if (e < E) { s = ei[e]; d = ei[E + e]; } else { s = d = (int)(e - E); }
  float norm = dinv[s] * dinv[d];
  const float* xs = xw + (size_t)s * 8;
  float* ad = agg + (size_t)d * 8;
#pragma unroll
  for (int j = 0; j < 8; ++j) atomicAdd(&ad[j], xs[j] * norm);
}

// ---------------- BN2 column stats (block-LDS reduce, then global atomics) ----------------
__global__ void k_stat2(const float* __restrict__ agg, float* __restrict__ stat2, int N) {
  __shared__ float ss[16];
  int tid = threadIdx.x;
  if (tid < 16) ss[tid] = 0.f;
  __syncthreads();
  int n = blockIdx.x * blockDim.x + tid;
  if (n < N) {
#pragma unroll
    for (int j = 0; j < 8; ++j) {
      float v = agg[(size_t)n * 8 + j];
      atomicAdd(&ss[j], v);
      atomicAdd(&ss[8 + j], v * v);
    }
  }
  __syncthreads();
  if (tid < 16) atomicAdd(&stat2[tid], ss[tid]);
}

// ---------------- fold BN2 into per-feature scale/shift (b_gcn cancels) ----------------
__global__ void k_fold2(const float* __restrict__ stat2, const float* __restrict__ g2,
                        const float* __restrict__ be2, float* __restrict__ fold2, int N) {
  int j = threadIdx.x;
  if (j >= 8) return;
  float mean = stat2[j] / (float)N;
  float var = stat2[8 + j] / (float)N - mean * mean;
  float s = g2[j] * rsqrtf(var + BN_EPS);
  fold2[j] = s;
  fold2[8 + j] = be2[j] - mean * s;
}

// ------- dense head GEMM: h1[B,32] = lrelu(bn2(agg).reshape(B,4096) @ W1 + b1) (WMMA) -------
__global__ void k_dense1(const float* __restrict__ agg, const float* __restrict__ W1,
                         const float* __restrict__ bias1, const float* __restrict__ fold2,
                         float* __restrict__ h1) {
  int lane = threadIdx.x & 31, ct = threadIdx.x >> 5;  // 2 waves: column tiles 0,1
  int hf = lane >> 4, l16 = lane & 15;
  int rt = blockIdx.x;                                 // 16-row tile of graphs
  int h2 = 2 * hf;
  // BN scale/shift for the 4 feature slots this lane touches (k mod 8 pattern)
  float sA0 = fold2[h2], sA1 = fold2[h2 + 1], sB0 = fold2[h2 + 4], sB1 = fold2[h2 + 5];
  float tA0 = fold2[8 + h2], tA1 = fold2[8 + h2 + 1], tB0 = fold2[8 + h2 + 4], tB1 = fold2[8 + h2 + 5];
  const float* arow = agg + (size_t)(rt * 16 + l16) * 4096;
  const float* wb = W1 + ct * 16 + l16;
  v8f c = {};
  for (int kk = 0; kk < 4096; kk += 8) {
    int ka = kk + h2;
    v2f a, b;
    a[0] = arow[ka] * sA0 + tA0;
    a[1] = arow[ka + 1] * sA1 + tA1;
    b[0] = wb[(size_t)ka * 32];
    b[1] = wb[(size_t)(ka + 1) * 32];
    c = __builtin_amdgcn_wmma_f32_16x16x4_f32(false, a, false, b, (short)0, c, false, false);
    int kb = kk + 4 + h2;
    a[0] = arow[kb] * sB0 + tB0;
    a[1] = arow[kb + 1] * sB1 + tB1;
    b[0] = wb[(size_t)kb * 32];
    b[1] = wb[(size_t)(kb + 1) * 32];
    c = __builtin_amdgcn_wmma_f32_16x16x4_f32(false, a, false, b, (short)0, c, false, false);
  }
  int col = ct * 16 + l16;
  float bv = bias1[col];
#pragma unroll
  for (int g = 0; g < 8; ++g) {
    int row = rt * 16 + g + 8 * hf;
    h1[row * 32 + col] = lrelu001(c[g] + bv);
  }
}

// ---------------- final tiny MLP chain -> (mu, logstd) ----------------
__global__ void k_head(const float* __restrict__ h1, const float* __restrict__ W2,
                       const float* __restrict__ b2, const float* __restrict__ Wt,
                       const float* __restrict__ bt, const float* __restrict__ Wmu,
                       const float* __restrict__ bmu, const float* __restrict__ Wls,
                       const float* __restrict__ bls, float* __restrict__ out, int B) {
  int g = blockIdx.x * blockDim.x + threadIdx.x;
  if (g >= B) return;
  float v1[32];
#pragma unroll
  for (int k = 0; k < 32; ++k) v1[k] = h1[(size_t)g * 32 + k];
  float v2[16];
#pragma unroll
  for (int j = 0; j < 16; ++j) {
    float acc = b2[j];
#pragma unroll
    for (int k = 0; k < 32; ++k) acc += v1[k] * W2[k * 16 + j];
    v2[j] = lrelu001(acc);
  }
  float v3[64];
#pragma unroll
  for (int j = 0; j < 64; ++j) {
    float acc = bt[j];
#pragma unroll
    for (int k = 0; k < 16; ++k) acc += v2[k] * Wt[k * 64 + j];
    v3[j] = lrelu001(acc);
  }
  for (int j = 0; j < 64; ++j) {
    float mu = bmu[j], ls = bls[j];
#pragma unroll
    for (int k = 0; k < 64; ++k) {
      mu += v3[k] * Wmu[k * 64 + j];
      ls += v3[k] * Wls[k * 64 + j];
    }
    out[(size_t)g * 64 + j] = mu;
    out[(size_t)(B + g) * 64 + j] = ls;
  }
}

extern "C" void kernel_launch(void* const* d_in, const int* in_sizes, int n_in,
                              void* d_out, int out_size, void* d_ws, size_t ws_size,
                              hipStream_t stream) {
  const float* x    = (const float*)d_in[0];
  const int*   ei   = (const int*)d_in[1];
  const float* Wgat = (const float*)d_in[2];
  const float* attS = (const float*)d_in[3];
  const float* attD = (const float*)d_in[4];
  // d_in[5] b_gat, d_in[7] b_ht, d_in[11] b_gcn cancel under BatchNorm (column-constant shifts)
  const float* Wht  = (const float*)d_in[6];
  const float* g1   = (const float*)d_in[8];
  const float* be1  = (const float*)d_in[9];
  const float* Wgcn = (const float*)d_in[10];
  const float* g2   = (const float*)d_in[12];
  const float* be2  = (const float*)d_in[13];
  const float* W1   = (const float*)d_in[14];
  const float* b1   = (const float*)d_in[15];
  const float* W2   = (const float*)d_in[16];
  const float* b2   = (const float*)d_in[17];
  const float* Wt   = (const float*)d_in[18];
  const float* bt   = (const float*)d_in[19];
  const float* Wmu  = (const float*)d_in[20];
  const float* bmu  = (const float*)d_in[21];
  const float* Wls  = (const float*)d_in[22];
  const float* bls  = (const float*)d_in[23];

  const size_t N = (size_t)in_sizes[0] / 32;      // 131072
  const long long E = (long long)in_sizes[1] / 2; // 2097152
  const long long Etot = E + (long long)N;        // + self loops
  const int B = (int)(N / 512);                   // 256 graphs

  float* ws = (float*)d_ws;
  size_t off = 0;
  float* H     = ws + off; off += N * 48;   // GAT features h (aliased later by HT/XW/DINV)
  float* ASRC  = ws + off; off += N * 3;
  float* ADST  = ws + off; off += N * 3;
  float* FOLD1 = ws + off; off += 136;
  float* FOLD2 = ws + off; off += 16;
  float* H1    = ws + off; off += (size_t)B * 32;
  float* ZB    = ws + off;                  // zero-init region start
  float* OUT1  = ws + off; off += N * 48;   // GAT aggregate (aliased later by AGG)
  unsigned* MORD = (unsigned*)(ws + off); off += N * 3;  // zero == ordered(-inf)
  float* Z     = ws + off; off += N * 3;
  float* DEG   = ws + off; off += N;
  float* STAT1 = ws + off; off += 32;
  float* STAT2 = ws + off; off += 16;
  size_t nzero = (size_t)((ws + off) - ZB);
  // aliases (lifetimes don't overlap across the stream-ordered kernel sequence)
  float* HT   = H;            // [N,16]
  float* XW   = H + N * 16;   // [N,8]
  float* DINV = H + N * 24;   // [N]
  float* AGG  = OUT1;         // [N,8], re-zeroed in k_xw

  int eblocks = (int)((Etot + 255) / 256);
  int nblocks = (int)((N + 255) / 256);

  k_zero<<<2048, 256, 0, stream>>>(ZB, nzero);
  k_gat_gemm<<<(int)(N / 128), 256, 0, stream>>>(x, Wgat, H, (int)N);
  k_att<<<nblocks, 256, 0, stream>>>(H, attS, attD, ASRC, ADST, (int)N);
  k_edge_max<<<eblocks, 256, 0, stream>>>(ei, ASRC, ADST, MORD, E, Etot);
  k_edge_z<<<eblocks, 256, 0, stream>>>(ei, ASRC, ADST, MORD, Z, DEG, E, Etot);
  k_edge_msg<<<eblocks, 256, 0, stream>>>(ei, ASRC, ADST, MORD, Z, H, OUT1, E, Etot);
  k_ht_gemm<<<(int)(N / 128), 256, 0, stream>>>(OUT1, Wht, HT, STAT1, (int)N);
  k_fold1<<<1, 128, 0, stream>>>(STAT1, g1, be1, Wgcn, FOLD1, (int)N);
  k_xw<<<nblocks, 256, 0, stream>>>(HT, FOLD1, DEG, XW, DINV, AGG, (int)N);
  k_gcn_edge<<<eblocks, 256, 0, stream>>>(ei, DINV, XW, AGG, E, Etot);
  k_stat2<<<nblocks, 256, 0, stream>>>(AGG, STAT2, (int)N);
  k_fold2<<<1, 32, 0, stream>>>(STAT2, g2, be2, FOLD2, (int)N);
  k_dense1<<<B / 16, 64, 0, stream>>>(AGG, W1, b1, FOLD2, H1);
  k_head<<<(B + 255) / 256, 256, 0, stream>>>(H1, W2, b2, Wt, bt, Wmu, bmu, Wls, bls,
                                              (float*)d_out, B);
}